// all_Centroid_32822140076442
// MI455X (gfx1250) — compile-verified
//
#include <hip/hip_runtime.h>
#include <hip/hip_bf16.h>

namespace {
constexpr int   DIM    = 128;
constexpr int   BK     = 2048 * 128;  // total (b,k) pairs
constexpr int   WPB    = 8;           // wave32s per block
constexpr int   TPB    = WPB * 32;    // 256 threads
constexpr int   CHUNK  = 32;          // contiguous pairs per wave
constexpr int   NBLK   = BK / (WPB * CHUNK);   // 1024 blocks -> 8192 waves
constexpr int   PPITCH = 68;          // scratch floats per pair: 8 slots * 8 + 4 pad
constexpr float EPSF   = 1e-5f;
constexpr float MAXNF  = 1.0f - 1e-5f;
}

// ---- CDNA5 async global->LDS DMA (gfx1250) --------------------------------
#if __has_builtin(__builtin_amdgcn_global_load_async_to_lds_b128)
#define HAVE_ASYNC 1
#else
#define HAVE_ASYNC 0
#endif

#if __has_builtin(__builtin_amdgcn_s_wait_asynccnt)
#define ASYNC_WAIT(n) __builtin_amdgcn_s_wait_asynccnt(n)
#elif HAVE_ASYNC
#define ASYNC_WAIT(n) asm volatile("s_wait_asynccnt %0" :: "i"(n) : "memory")
#else
#define ASYNC_WAIT(n) do {} while (0)
#endif

typedef int v4i __attribute__((ext_vector_type(4)));
typedef __attribute__((address_space(1))) v4i* gv4p;  // global int4*
typedef __attribute__((address_space(3))) v4i* lv4p;  // LDS int4*

__device__ __forceinline__ void stage16(float* lds_dst, const float* g_src) {
#if HAVE_ASYNC
  __builtin_amdgcn_global_load_async_to_lds_b128(
      (gv4p)g_src, (lv4p)lds_dst, /*imm offset=*/0, /*cpol=*/0);
#else
  *(float4*)lds_dst = *(const float4*)g_src;
#endif
}

__global__ __launch_bounds__(TPB) void poincare_score_kernel(
    const float* __restrict__ Eh,  const float* __restrict__ rvh,
    const float* __restrict__ Wm,  const float* __restrict__ bias0,
    const float* __restrict__ bias1,
    const int* __restrict__ hidx, const int* __restrict__ ridx,
    const int* __restrict__ tidx, float* __restrict__ out)
{
  // per wave: 2-deep staging buffer (4 rows x 512B) + per-pair partial scratch
  __shared__ __align__(16) float stage[WPB][2][4][DIM];        // 32 KB/block
  __shared__ __align__(16) float scratch[WPB][CHUNK][PPITCH];  // ~68 KB/block

  const int lane = threadIdx.x & 31;
  const int wv   = threadIdx.x >> 5;
  const int wid  = blockIdx.x * WPB + wv;
  const int base = wid * CHUNK;            // 32 contiguous pairs for this wave
  const int off4 = lane * 4;

  // lane i's value == pair (base+i)'s index: feeds DMA addr gen (readlane)
  // in phase 1 and the bias gathers / store in phase 2.
  const int vh = hidx[base + lane];
  const int vr = ridx[base + lane];
  const int vt = tidx[base + lane];

  // ---------------- Phase 1: DMA-pipelined partial reductions --------------
  {
    const int h0 = __builtin_amdgcn_readlane(vh, 0);
    const int r0 = __builtin_amdgcn_readlane(vr, 0);
    const int t0 = __builtin_amdgcn_readlane(vt, 0);
    stage16(&stage[wv][0][0][off4], Eh  + (size_t)h0 * DIM + off4);
    stage16(&stage[wv][0][1][off4], Eh  + (size_t)t0 * DIM + off4);
    stage16(&stage[wv][0][2][off4], rvh + (size_t)r0 * DIM + off4);
    stage16(&stage[wv][0][3][off4], Wm  + (size_t)r0 * DIM + off4);
  }

  int buf = 0;
#pragma unroll 1
  for (int i = 0; i < CHUNK; ++i) {
    if (i < CHUNK - 1) {                   // prefetch next pair
      const int hn = __builtin_amdgcn_readlane(vh, i + 1);
      const int rn = __builtin_amdgcn_readlane(vr, i + 1);
      const int tn = __builtin_amdgcn_readlane(vt, i + 1);
      const int nb = buf ^ 1;
      stage16(&stage[wv][nb][0][off4], Eh  + (size_t)hn * DIM + off4);
      stage16(&stage[wv][nb][1][off4], Eh  + (size_t)tn * DIM + off4);
      stage16(&stage[wv][nb][2][off4], rvh + (size_t)rn * DIM + off4);
      stage16(&stage[wv][nb][3][off4], Wm  + (size_t)rn * DIM + off4);
      ASYNC_WAIT(4);                       // current buffer's 4 DMAs done
    } else {
      ASYNC_WAIT(0);
    }
    asm volatile("" ::: "memory");

    const float4 h4 = *(const float4*)&stage[wv][buf][0][off4];
    const float4 t4 = *(const float4*)&stage[wv][buf][1][off4];
    const float4 r4 = *(const float4*)&stage[wv][buf][2][off4];
    const float4 w4 = *(const float4*)&stage[wv][buf][3][off4];

    // s1=Σh², s2=Σt², s3=Σr², s4=Σt·r, s5=Σ(hw)², s6=Σhw·t, s7=Σhw·r
    float s1 = 0, s2 = 0, s3 = 0, s4 = 0, s5 = 0, s6 = 0, s7 = 0;
    const float hh[4] = {h4.x, h4.y, h4.z, h4.w};
    const float tt[4] = {t4.x, t4.y, t4.z, t4.w};
    const float rr[4] = {r4.x, r4.y, r4.z, r4.w};
    const float ww[4] = {w4.x, w4.y, w4.z, w4.w};
#pragma unroll
    for (int j = 0; j < 4; ++j) {
      const float h = hh[j], t = tt[j], r = rr[j], w = ww[j];
      s1 = fmaf(h, h, s1);
      s2 = fmaf(t, t, s2);
      s3 = fmaf(r, r, s3);
      s4 = fmaf(t, r, s4);
      const float hw = h * w;
      s5 = fmaf(hw, hw, s5);
      s6 = fmaf(hw, t, s6);
      s7 = fmaf(hw, r, s7);
    }
    // 2 butterfly levels -> lanes 0..7 hold 8 disjoint coset partials
#pragma unroll
    for (int m = 16; m >= 8; m >>= 1) {
      s1 += __shfl_xor(s1, m, 32); s2 += __shfl_xor(s2, m, 32);
      s3 += __shfl_xor(s3, m, 32); s4 += __shfl_xor(s4, m, 32);
      s5 += __shfl_xor(s5, m, 32); s6 += __shfl_xor(s6, m, 32);
      s7 += __shfl_xor(s7, m, 32);
    }
    if (lane < 8) {
      float* dst = &scratch[wv][i][lane * 8];
      *(float4*)(dst)     = make_float4(s1, s2, s3, s4);
      *(float4*)(dst + 4) = make_float4(s5, s6, s7, 0.0f);
    }
    buf ^= 1;
  }

  asm volatile("" ::: "memory");

  // ---------------- Phase 2: lane j finishes pair (base+j) ----------------
  float a1 = 0, a2 = 0, a3 = 0, a4 = 0, a5 = 0, a6 = 0, a7 = 0;
#pragma unroll
  for (int l = 0; l < 8; ++l) {
    const float* src = &scratch[wv][lane][l * 8];
    const float4 u = *(const float4*)(src);
    const float4 v = *(const float4*)(src + 4);
    a1 += u.x; a2 += u.y; a3 += u.z; a4 += u.w;
    a5 += v.x; a6 += v.y; a7 += v.z;
  }

  // ---- head branch: project -> log map -> *w -> exp map -> project ----
  const float n_h  = sqrtf(a1 + 1e-15f);
  const float sh   = (n_h > MAXNF) ? (MAXNF / n_h) : 1.0f;
  float nl         = sqrtf(sh * sh * a1 + 1e-15f);
  nl               = fminf(fmaxf(nl, EPSF), MAXNF);
  const float atnh = 0.5f * logf((1.0f + nl) / (1.0f - nl));
  const float alog = atnh / nl * sh;        // v = alog * h * w
  const float v2   = alog * alog * a5;      // Σv²
  float nv         = sqrtf(v2 + 1e-15f);
  nv               = fmaxf(nv, EPSF);
  const float aexp = tanhf(nv) / nv;        // exph = aexp * v
  const float e2   = aexp * aexp * v2;      // Σexph²
  const float ne   = sqrtf(e2 + 1e-15f);
  const float she  = (ne > MAXNF) ? (MAXNF / ne) : 1.0f;
  const float Ch   = she * aexp * alog;     // head = Ch * h * w
  const float head2 = Ch * Ch * a5;

  // ---- tail branch: project t, r -> Mobius add -> project ----
  const float n_t = sqrtf(a2 + 1e-15f);
  const float st  = (n_t > MAXNF) ? (MAXNF / n_t) : 1.0f;
  const float n_r = sqrtf(a3 + 1e-15f);
  const float sr  = (n_r > MAXNF) ? (MAXNF / n_r) : 1.0f;
  const float x2  = st * st * a2;           // Σ t_proj²
  const float y2  = sr * sr * a3;           // Σ r_proj²
  const float xy  = st * sr * a4;           // Σ t_proj·r_proj
  const float A   = 1.0f + 2.0f * xy + y2;
  const float Bc  = 1.0f - x2;
  float den       = 1.0f + 2.0f * xy + x2 * y2;
  den             = fmaxf(den, EPSF);
  const float tv2 = (A * A * x2 + 2.0f * A * Bc * xy + Bc * Bc * y2) / (den * den);
  const float ntv = sqrtf(tv2 + 1e-15f);
  const float stp = (ntv > MAXNF) ? (MAXNF / ntv) : 1.0f;
  const float tail2 = stp * stp * tv2;

  // ---- Poincare distance ----
  const float ht   = Ch * stp / den * (A * st * a6 + Bc * sr * a7);
  const float d2   = head2 + tail2 - 2.0f * ht;
  const float dden = fmaxf((1.0f - head2) * (1.0f - tail2), EPSF);
  float arg        = 1.0f + 2.0f * d2 / dden;
  arg              = fmaxf(arg, 1.0f + 1e-7f);
  const float dist = acoshf(arg);

  // lane's own vh/vt are exactly this pair's indices -> gather biases
  out[base + lane] = bias0[vh] + bias1[vt] - dist;
}

extern "C" void kernel_launch(void* const* d_in, const int* in_sizes, int n_in,
                              void* d_out, int out_size, void* d_ws, size_t ws_size,
                              hipStream_t stream) {
  (void)in_sizes; (void)n_in; (void)out_size; (void)d_ws; (void)ws_size;
  const float* Eh  = (const float*)d_in[0];
  const float* rvh = (const float*)d_in[1];
  const float* Wm  = (const float*)d_in[2];
  const float* b0  = (const float*)d_in[3];
  const float* b1  = (const float*)d_in[4];
  const int*   hi  = (const int*)d_in[5];
  const int*   ri  = (const int*)d_in[6];
  const int*   ti  = (const int*)d_in[7];
  float*       out = (float*)d_out;
  poincare_score_kernel<<<dim3(NBLK), dim3(TPB), 0, stream>>>(
      Eh, rvh, Wm, b0, b1, hi, ri, ti, out);
}